// GCN_47888885350563
// MI455X (gfx1250) — compile-verified
//
#include <hip/hip_runtime.h>
#include <math.h>

#define N_NODES 100000
#define N_EDGES 1600000
#define NF 256      // input features
#define NH 128      // hidden
#define NC 40       // classes
#define NCP 48      // classes padded to multiple of 16 for WMMA

typedef float v2f __attribute__((ext_vector_type(2)));
typedef float v8f __attribute__((ext_vector_type(8)));

// ---------------- degree / norm ----------------
__global__ void k_degree(const int* __restrict__ dst, float* __restrict__ deg) {
    int e = blockIdx.x * blockDim.x + threadIdx.x;
    if (e < N_EDGES) atomicAdd(&deg[dst[e]], 1.0f);
}

__global__ void k_dinv(float* __restrict__ deg) {
    int i = blockIdx.x * blockDim.x + threadIdx.x;
    if (i < N_NODES) deg[i] = rsqrtf(deg[i] + 1.0f);  // +1 self loop
}

// ---------------- fp32 WMMA GEMM: C[M,LDB] = A[M,K(LDA)] * B[K,LDB] ----------------
// grid.x = M/16 tiles; blockDim.x = (LDB/16)*32, one wave per 16x16 N-tile.
template<int K, int LDA, int LDB>
__global__ void k_gemm_wmma(const float* __restrict__ A, const float* __restrict__ B,
                            float* __restrict__ C) {
    const int mt   = blockIdx.x;
    const int nt   = threadIdx.x >> 5;   // wave id == N tile
    const int lane = threadIdx.x & 31;
    const int half = lane >> 4;          // K-half select
    const int lm   = lane & 15;          // M (for A) / N (for B,C)

    const float* arow = A + (size_t)(mt * 16 + lm) * LDA;
    const float* bcol = B + nt * 16 + lm;

    v8f c = {0.f, 0.f, 0.f, 0.f, 0.f, 0.f, 0.f, 0.f};
    #pragma unroll 4
    for (int kb = 0; kb < K; kb += 4) {
        const int ka = kb + 2 * half;
        v2f a; a.x = arow[ka];                 a.y = arow[ka + 1];
        v2f b; b.x = bcol[(size_t)ka * LDB];   b.y = bcol[(size_t)(ka + 1) * LDB];
        c = __builtin_amdgcn_wmma_f32_16x16x4_f32(false, a, false, b,
                                                  (short)0, c, false, false);
    }
    float* out = C + (size_t)(mt * 16 + 8 * half) * LDB + nt * 16 + lm;
    #pragma unroll
    for (int v = 0; v < 8; ++v) out[(size_t)v * LDB] = c[v];
}

// ---------------- layer 1: self loop init, edge scatter, bias+relu ----------------
__global__ void k_self1(const float* __restrict__ xw, const float* __restrict__ dinv,
                        float* __restrict__ agg) {
    int id = blockIdx.x * blockDim.x + threadIdx.x;
    if (id < N_NODES * NH) {
        int i = id >> 7;              // / NH
        float di = dinv[i];
        agg[id] = xw[id] * di * di;
    }
}

__global__ void k_edge1(const int* __restrict__ src, const int* __restrict__ dst,
                        const float* __restrict__ dinv, const float* __restrict__ xw,
                        float* __restrict__ agg) {
    int gid  = blockIdx.x * blockDim.x + threadIdx.x;
    int e    = gid >> 5;
    if (e >= N_EDGES) return;
    int lane = gid & 31;
    int s = src[e], d = dst[e];
    float norm = dinv[s] * dinv[d];
    const float4 v = *(const float4*)(xw + (size_t)s * NH + lane * 4);
    float* o = agg + (size_t)d * NH + lane * 4;
    atomicAdd(o + 0, v.x * norm);
    atomicAdd(o + 1, v.y * norm);
    atomicAdd(o + 2, v.z * norm);
    atomicAdd(o + 3, v.w * norm);
}

__global__ void k_relu_bias(float* __restrict__ agg, const float* __restrict__ b) {
    int id = blockIdx.x * blockDim.x + threadIdx.x;
    if (id < N_NODES * NH) agg[id] = fmaxf(agg[id] + b[id & (NH - 1)], 0.f);
}

// ---------------- layer 2 ----------------
__global__ void k_padw2(const float* __restrict__ W2, float* __restrict__ W2p) {
    int id = blockIdx.x * blockDim.x + threadIdx.x;
    if (id < NH * NCP) {
        int k = id / NCP, n = id % NCP;
        W2p[id] = (n < NC) ? W2[k * NC + n] : 0.f;
    }
}

__global__ void k_self2(const float* __restrict__ xw2, const float* __restrict__ dinv,
                        float* __restrict__ agg2) {
    int id = blockIdx.x * blockDim.x + threadIdx.x;
    if (id < N_NODES * NC) {
        int i = id / NC, f = id - i * NC;
        float di = dinv[i];
        agg2[id] = xw2[(size_t)i * NCP + f] * di * di;
    }
}

__global__ void k_edge2(const int* __restrict__ src, const int* __restrict__ dst,
                        const float* __restrict__ dinv, const float* __restrict__ xw2,
                        float* __restrict__ agg2) {
    int gid  = blockIdx.x * blockDim.x + threadIdx.x;
    int e    = gid >> 5;
    if (e >= N_EDGES) return;
    int lane = gid & 31;
    int s = src[e], d = dst[e];
    float norm = dinv[s] * dinv[d];
    atomicAdd(agg2 + (size_t)d * NC + lane, xw2[(size_t)s * NCP + lane] * norm);
    if (lane < NC - 32)
        atomicAdd(agg2 + (size_t)d * NC + lane + 32,
                  xw2[(size_t)s * NCP + lane + 32] * norm);
}

__global__ void k_bias2(const float* __restrict__ agg2, const float* __restrict__ b2,
                        float* __restrict__ logits) {
    int id = blockIdx.x * blockDim.x + threadIdx.x;
    if (id < N_NODES * NC) {
        int f = id % NC;
        logits[id] = agg2[id] + b2[f];
    }
}

// ---------------- softmax over axis 0 (columns): per-column max & sum ----------------
__global__ void k_colstats(const float* __restrict__ logits,
                           float* __restrict__ colmax, float* __restrict__ colsum) {
    __shared__ float red[256];
    int c = blockIdx.x;                  // 0..NC-1
    float m = -INFINITY;
    for (int i = threadIdx.x; i < N_NODES; i += blockDim.x)
        m = fmaxf(m, logits[(size_t)i * NC + c]);
    red[threadIdx.x] = m; __syncthreads();
    for (int s = 128; s > 0; s >>= 1) {
        if (threadIdx.x < s) red[threadIdx.x] = fmaxf(red[threadIdx.x], red[threadIdx.x + s]);
        __syncthreads();
    }
    float cm = red[0]; __syncthreads();
    float acc = 0.f;
    for (int i = threadIdx.x; i < N_NODES; i += blockDim.x)
        acc += expf(logits[(size_t)i * NC + c] - cm);
    red[threadIdx.x] = acc; __syncthreads();
    for (int s = 128; s > 0; s >>= 1) {
        if (threadIdx.x < s) red[threadIdx.x] += red[threadIdx.x + s];
        __syncthreads();
    }
    if (threadIdx.x == 0) { colmax[c] = cm; colsum[c] = red[0]; }
}

// ---------------- final: log_softmax(rows) + softmax(cols); logits already in d_out ----
__global__ void k_finalout(const float* __restrict__ logits,
                           const float* __restrict__ colmax, const float* __restrict__ colsum,
                           float* __restrict__ out_lsm, float* __restrict__ out_sm0) {
    int gid  = blockIdx.x * blockDim.x + threadIdx.x;
    int node = gid >> 5;
    if (node >= N_NODES) return;
    int lane = gid & 31;
    const float* row = logits + (size_t)node * NC;
    float v0 = row[lane];
    bool has2 = lane < (NC - 32);
    float v1 = has2 ? row[lane + 32] : -INFINITY;
    float m = fmaxf(v0, v1);
    #pragma unroll
    for (int off = 16; off > 0; off >>= 1) m = fmaxf(m, __shfl_xor(m, off, 32));
    float s = expf(v0 - m) + (has2 ? expf(v1 - m) : 0.f);
    #pragma unroll
    for (int off = 16; off > 0; off >>= 1) s += __shfl_xor(s, off, 32);
    float lse = m + logf(s);
    out_lsm[(size_t)node * NC + lane] = v0 - lse;
    out_sm0[(size_t)node * NC + lane] = expf(v0 - colmax[lane]) / colsum[lane];
    if (has2) {
        out_lsm[(size_t)node * NC + lane + 32] = v1 - lse;
        out_sm0[(size_t)node * NC + lane + 32] =
            expf(v1 - colmax[lane + 32]) / colsum[lane + 32];
    }
}

extern "C" void kernel_launch(void* const* d_in, const int* in_sizes, int n_in,
                              void* d_out, int out_size, void* d_ws, size_t ws_size,
                              hipStream_t stream) {
    (void)in_sizes; (void)n_in; (void)out_size; (void)ws_size;
    const float* x   = (const float*)d_in[0];
    const int*   ei  = (const int*)d_in[1];
    const float* W1  = (const float*)d_in[2];
    const float* b1  = (const float*)d_in[3];
    const float* W2  = (const float*)d_in[4];
    const float* b2  = (const float*)d_in[5];
    const int* src = ei;
    const int* dst = ei + N_EDGES;

    float* ws = (float*)d_ws;
    size_t off = 0;
    float* dinv = ws + off; off += N_NODES; off = (off + 3) & ~(size_t)3;
    float* xw1  = ws + off; off += (size_t)N_NODES * NH;
    float* agg1 = ws + off; off += (size_t)N_NODES * NH;   // becomes h after relu
    float* W2p  = ws + off; off += NH * NCP;
    float* xw2  = ws + off; off += (size_t)N_NODES * NCP;
    float* agg2 = ws + off; off += (size_t)N_NODES * NC;
    float* colmax = ws + off; off += 64;
    float* colsum = ws + off; off += 64;

    float* out_lsm = (float*)d_out;
    float* logits  = (float*)d_out + (size_t)N_NODES * NC;
    float* out_sm0 = (float*)d_out + 2 * (size_t)N_NODES * NC;

    // degree + normalization
    hipMemsetAsync(dinv, 0, N_NODES * sizeof(float), stream);
    k_degree<<<(N_EDGES + 255) / 256, 256, 0, stream>>>(dst, dinv);
    k_dinv<<<(N_NODES + 255) / 256, 256, 0, stream>>>(dinv);

    // layer 1
    k_gemm_wmma<NF, NF, NH><<<N_NODES / 16, (NH / 16) * 32, 0, stream>>>(x, W1, xw1);
    k_self1<<<(N_NODES * NH + 255) / 256, 256, 0, stream>>>(xw1, dinv, agg1);
    k_edge1<<<(size_t)N_EDGES * 32 / 256, 256, 0, stream>>>(src, dst, dinv, xw1, agg1);
    k_relu_bias<<<(N_NODES * NH + 255) / 256, 256, 0, stream>>>(agg1, b1);

    // layer 2
    k_padw2<<<(NH * NCP + 255) / 256, 256, 0, stream>>>(W2, W2p);
    k_gemm_wmma<NH, NH, NCP><<<N_NODES / 16, (NCP / 16) * 32, 0, stream>>>(agg1, W2p, xw2);
    k_self2<<<(N_NODES * NC + 255) / 256, 256, 0, stream>>>(xw2, dinv, agg2);
    k_edge2<<<(size_t)N_EDGES * 32 / 256, 256, 0, stream>>>(src, dst, dinv, xw2, agg2);
    k_bias2<<<(N_NODES * NC + 255) / 256, 256, 0, stream>>>(agg2, b2, logits);

    // outputs
    k_colstats<<<NC, 256, 0, stream>>>(logits, colmax, colsum);
    k_finalout<<<(N_NODES * 32 + 255) / 256, 256, 0, stream>>>(logits, colmax, colsum,
                                                               out_lsm, out_sm0);
}